// HierarchicalTensorRefinement_90426241450250
// MI455X (gfx1250) — compile-verified
//
#include <hip/hip_runtime.h>

typedef float v2f __attribute__((ext_vector_type(2)));
typedef float v4f __attribute__((ext_vector_type(4)));
typedef float v8f __attribute__((ext_vector_type(8)));

#define NB 2
#define NN 4096
#define NM 32
#define ND 128
#define NE 64
#define LN_EPS 1e-5f

// ---------------------------------------------------------------------------
// Kernel 1: per-(b,n) projections.
// q-record[e][0..2] = x1 . to_queries  (ml = 3)
// q-record[e][3..7] = x2 . to_queries  (ml = 5)
// k-record[e][0..2] = x1 . to_keys1
// k-record[e][3..7] = x2 . to_keys2
// 64 threads per (b,n), one e per thread; 4 (b,n) per 256-thread block.
// ---------------------------------------------------------------------------
__global__ __launch_bounds__(256) void proj_kernel(
    const float* __restrict__ x1, const float* __restrict__ x2,
    const float* __restrict__ Wq, const float* __restrict__ Wk1,
    const float* __restrict__ Wk2,
    float* __restrict__ Q, float* __restrict__ K)
{
    const int group = threadIdx.x >> 6;          // 0..3
    const int e     = threadIdx.x & 63;
    const long bn   = (long)blockIdx.x * 4 + group;   // 0..B*N-1

    const float* x1p = x1 + bn * (ND * 3);
    const float* x2p = x2 + bn * (ND * 5);

    float qa[8] = {0.f,0.f,0.f,0.f,0.f,0.f,0.f,0.f};
    float ka[8] = {0.f,0.f,0.f,0.f,0.f,0.f,0.f,0.f};

    for (int d = 0; d < ND; ++d) {
        const float wq = Wq [d * NE + e];
        const float w1 = Wk1[d * NE + e];
        const float w2 = Wk2[d * NE + e];
        const float a0 = x1p[d*3+0], a1 = x1p[d*3+1], a2 = x1p[d*3+2];
        qa[0] += wq*a0; qa[1] += wq*a1; qa[2] += wq*a2;
        ka[0] += w1*a0; ka[1] += w1*a1; ka[2] += w1*a2;
        const float b0 = x2p[d*5+0], b1 = x2p[d*5+1], b2 = x2p[d*5+2];
        const float b3 = x2p[d*5+3], b4 = x2p[d*5+4];
        qa[3] += wq*b0; qa[4] += wq*b1; qa[5] += wq*b2; qa[6] += wq*b3; qa[7] += wq*b4;
        ka[3] += w2*b0; ka[4] += w2*b1; ka[5] += w2*b2; ka[6] += w2*b3; ka[7] += w2*b4;
    }

    float* qrec = Q + (bn * NE + e) * 8;
    float* krec = K + (bn * NE + e) * 8;
    v4f q0 = {qa[0], qa[1], qa[2], qa[3]};
    v4f q1 = {qa[4], qa[5], qa[6], qa[7]};
    v4f k0 = {ka[0], ka[1], ka[2], ka[3]};
    v4f k1 = {ka[4], ka[5], ka[6], ka[7]};
    *(v4f*)(qrec)     = q0;
    *(v4f*)(qrec + 4) = q1;
    *(v4f*)(krec)     = k0;
    *(v4f*)(krec + 4) = k1;
}

// ---------------------------------------------------------------------------
// Kernel 2: fused gather + inner-product + LayerNorm + dual WMMA GEMM.
// One 256-thread workgroup (8 wave32) per (b,n).
//   w_tile [32][128]  = neighbor inner products (concat of ip1|ip2), then LN.
//   edge  = sigmoid(w_tile @ edge_w)   via v_wmma_f32_16x16x4_f32
//   res   = t_tile @ res_w
//   out   = edge + res
// Each wave owns one 16-wide N stripe and runs ONE K-loop computing 4
// independent accumulator chains (edge/res x M-tile 0/1), sharing the
// B fragments between M-tiles.
// ---------------------------------------------------------------------------
__global__ __launch_bounds__(256) void fused_kernel(
    const float* __restrict__ t_ij, const int* __restrict__ nbr,
    const float* __restrict__ Q, const float* __restrict__ K,
    const float* __restrict__ lnw, const float* __restrict__ edge_w,
    const float* __restrict__ res_w, float* __restrict__ out)
{
    __shared__ float sQ[NE * 8];          // this node's q record
    __shared__ float sW[NM][ND + 4];      // w_ij tile (padded vs bank conflicts)
    __shared__ float sT[NM][ND + 4];      // t_ij tile
    __shared__ int   sIdx[NM];

    const long bn  = blockIdx.x;          // 0..B*N-1
    const int  b   = (int)(bn >> 12);     // N == 4096
    const int  tid = threadIdx.x;

    // Pull the (L2-resident) weight matrices toward this WGP while the
    // gather / LN phases run: 2 cachelines per thread per matrix.
    __builtin_prefetch(edge_w + (size_t)tid * 64,       0, 0);
    __builtin_prefetch(edge_w + (size_t)tid * 64 + 32,  0, 0);
    __builtin_prefetch(res_w  + (size_t)tid * 64,       0, 0);
    __builtin_prefetch(res_w  + (size_t)tid * 64 + 32,  0, 0);

    // ---- stage q record, neighbor indices, t tile -------------------------
    for (int i = tid; i < NE * 8; i += 256) sQ[i] = Q[bn * NE * 8 + i];
    if (tid < NM) sIdx[tid] = nbr[bn * NM + tid];

    const float* tptr = t_ij + bn * (NM * ND);
    for (int i = tid; i < (NM * ND) / 4; i += 256) {
        v4f v = ((const v4f*)tptr)[i];
        const int j = (i * 4) / ND, c = (i * 4) % ND;
        sT[j][c] = v.x; sT[j][c+1] = v.y; sT[j][c+2] = v.z; sT[j][c+3] = v.w;
    }
    __syncthreads();

    // ---- gather + inner products: thread -> (j = tid/8, 8 e-values) -------
    {
        const int  j  = tid >> 3;
        const int  e0 = (tid & 7) * 8;
        const long nb = (long)b * NN + sIdx[j];
        const float* krec = K + (nb * NE + e0) * 8;
        #pragma unroll
        for (int ee = 0; ee < 8; ++ee) {
            const int e = e0 + ee;
            const float* kr = krec + ee * 8;
            const float* qr = sQ + e * 8;
            const float s1 = qr[0]*kr[0] + qr[1]*kr[1] + qr[2]*kr[2];
            const float s2 = qr[3]*kr[3] + qr[4]*kr[4] + qr[5]*kr[5]
                           + qr[6]*kr[6] + qr[7]*kr[7];
            sW[j][e]      = s1;   // ip1 -> first 64 features
            sW[j][NE + e] = s2;   // ip2 -> last 64 features
        }
    }
    __syncthreads();

    // ---- LayerNorm(128) per row, weight-only; wave32 shfl reduction -------
    {
        const int wv = tid >> 5, lane = tid & 31;
        for (int r = 0; r < 4; ++r) {
            const int j = wv * 4 + r;
            float s = 0.f, ss = 0.f;
            #pragma unroll
            for (int c = lane; c < ND; c += 32) {
                const float v = sW[j][c];
                s += v; ss += v * v;
            }
            #pragma unroll
            for (int off = 16; off > 0; off >>= 1) {
                s  += __shfl_xor(s,  off, 32);
                ss += __shfl_xor(ss, off, 32);
            }
            const float mu  = s * (1.f / ND);
            const float var = ss * (1.f / ND) - mu * mu;
            const float inv = rsqrtf(var + LN_EPS);
            #pragma unroll
            for (int c = lane; c < ND; c += 32)
                sW[j][c] = (sW[j][c] - mu) * inv * lnw[c];
        }
    }
    __syncthreads();

    // ---- fused dual GEMM via V_WMMA_F32_16X16X4_F32 -----------------------
    // Fragment layout per ISA: lanes<16 hold K={0,1}, lanes>=16 K={2,3};
    // D: VGPR r -> M = r + (lane>=16 ? 8 : 0), N = lane%16.
    const int wv    = tid >> 5;               // N-stripe (0..7)
    const int lane  = tid & 31;
    const int lhalf = (lane >= 16) ? 2 : 0;
    const int lmod  = lane & 15;
    const int arow0 = lmod;                   // M-tile 0 rows
    const int arow1 = 16 + lmod;              // M-tile 1 rows
    const int bcol  = wv * 16 + lmod;
    float* outp = out + bn * (NM * ND);

    v8f acc_e0 = {}, acc_e1 = {}, acc_r0 = {}, acc_r1 = {};

    #pragma unroll 4
    for (int kk = 0; kk < ND / 4; ++kk) {
        const int kb = kk * 4 + lhalf;
        v2f be, br, a0, a1, t0, t1;
        be.x = edge_w[kb * ND + bcol];  be.y = edge_w[(kb + 1) * ND + bcol];
        br.x = res_w [kb * ND + bcol];  br.y = res_w [(kb + 1) * ND + bcol];
        a0.x = sW[arow0][kb];  a0.y = sW[arow0][kb + 1];
        a1.x = sW[arow1][kb];  a1.y = sW[arow1][kb + 1];
        t0.x = sT[arow0][kb];  t0.y = sT[arow0][kb + 1];
        t1.x = sT[arow1][kb];  t1.y = sT[arow1][kb + 1];
        // 4 independent accumulator chains; B fragments shared across M-tiles.
        acc_e0 = __builtin_amdgcn_wmma_f32_16x16x4_f32(
            false, a0, false, be, (short)0, acc_e0, false, false);
        acc_e1 = __builtin_amdgcn_wmma_f32_16x16x4_f32(
            false, a1, false, be, (short)0, acc_e1, false, false);
        acc_r0 = __builtin_amdgcn_wmma_f32_16x16x4_f32(
            false, t0, false, br, (short)0, acc_r0, false, false);
        acc_r1 = __builtin_amdgcn_wmma_f32_16x16x4_f32(
            false, t1, false, br, (short)0, acc_r1, false, false);
    }

    // ---- epilogue: out = sigmoid(edge) + res ------------------------------
    const int rhalf = (lane >= 16) ? 8 : 0;
    #pragma unroll
    for (int r = 0; r < 8; ++r) {
        const int row0 = r + rhalf;           // M-tile 0
        const int row1 = 16 + r + rhalf;      // M-tile 1
        const float e0 = 1.f / (1.f + __expf(-acc_e0[r]));
        const float e1 = 1.f / (1.f + __expf(-acc_e1[r]));
        outp[row0 * ND + bcol] = e0 + acc_r0[r];
        outp[row1 * ND + bcol] = e1 + acc_r1[r];
    }
}

// ---------------------------------------------------------------------------
extern "C" void kernel_launch(void* const* d_in, const int* in_sizes, int n_in,
                              void* d_out, int out_size, void* d_ws, size_t ws_size,
                              hipStream_t stream) {
    (void)in_sizes; (void)n_in; (void)out_size; (void)ws_size;

    const float* t_ij   = (const float*)d_in[0];
    const float* x1     = (const float*)d_in[1];
    const float* x2     = (const float*)d_in[2];
    const int*   nbr    = (const int*)  d_in[3];
    const float* Wq     = (const float*)d_in[4];
    const float* Wk1    = (const float*)d_in[5];
    const float* Wk2    = (const float*)d_in[6];
    const float* lnw    = (const float*)d_in[7];
    const float* edge_w = (const float*)d_in[8];
    const float* res_w  = (const float*)d_in[9];
    float* out = (float*)d_out;

    const long n_bn = (long)NB * NN;                  // 8192
    float* Q = (float*)d_ws;                          // [bn][e][8]
    float* K = Q + n_bn * NE * 8;                     // +16 MB

    proj_kernel<<<(int)(n_bn / 4), 256, 0, stream>>>(x1, x2, Wq, Wk1, Wk2, Q, K);
    fused_kernel<<<(int)n_bn, 256, 0, stream>>>(t_ij, nbr, Q, K, lnw,
                                                edge_w, res_w, out);
}